// Convpass_77756087927538
// MI455X (gfx1250) — compile-verified
//
#include <hip/hip_runtime.h>

typedef __attribute__((ext_vector_type(2))) float v2f;
typedef __attribute__((ext_vector_type(8))) float v8f;

#define DIMK 8
#define CCH 192
#define NTOK 198
#define NPAD 208   // 13 * 16
#define MT 13
#define NT 12

__device__ __forceinline__ float qgelu(float v) {
    // x * sigmoid(1.702 x)
    return v / (1.0f + __expf(-1.702f * v));
}

__global__ __launch_bounds__(256) void convpass_fused(
    const float* __restrict__ x,        // [B, 198, 192]
    const float* __restrict__ w_down,   // [8, 192]
    const float* __restrict__ b_down,   // [8]
    const float* __restrict__ conv_w,   // [8, 8, 3, 3]
    const float* __restrict__ conv_b,   // [8]
    const float* __restrict__ w_up,     // [192, 8]
    const float* __restrict__ b_up,     // [192]
    float* __restrict__ out)            // [B, 198, 192]
{
    __shared__ float s_wdT[CCH][16];      // w_down^T, cols 8..15 zero (N-mask for WMMA)
    __shared__ float s_wup[CCH][9];       // w_up[c][k], stride 9 -> conflict-free
    __shared__ float s_conv[DIMK * DIMK * 9];
    __shared__ float s_cb[DIMK];
    __shared__ float s_bd[DIMK];
    __shared__ float s_bu[CCH];
    __shared__ float s_d[NPAD][DIMK];     // post-downproj activations
    __shared__ float s_d2[NPAD][9];       // post-conv activations, stride 9

    const int tid  = threadIdx.x;
    const int b    = blockIdx.x;
    const int lane = tid & 31;
    const int wid  = tid >> 5;
    const int half = lane >> 4;   // lane group 0/1 (ISA WMMA layout)
    const int l16  = lane & 15;

    // ---- stage weights/biases into LDS ----
    for (int e = tid; e < CCH * 16; e += 256) {       // zero-padded w_down^T
        int c = e >> 4, k = e & 15;
        s_wdT[c][k] = (k < DIMK) ? w_down[k * CCH + c] : 0.0f;
    }
    for (int e = tid; e < CCH * DIMK; e += 256) {     // w_up is [c][k] row-major already
        s_wup[e >> 3][e & 7] = w_up[e];
    }
    for (int e = tid; e < DIMK * DIMK * 9; e += 256) s_conv[e] = conv_w[e];
    for (int e = tid; e < CCH; e += 256) s_bu[e] = b_up[e];
    if (tid < DIMK) { s_cb[tid] = conv_b[tid]; s_bd[tid] = b_down[tid]; }
    __syncthreads();

    // ---- Phase 1: down projection via V_WMMA_F32_16X16X4_F32, + bias + QuickGELU ----
    const float* xb = x + (size_t)b * NTOK * CCH;
    for (int t = wid; t < MT; t += 8) {
        const int n0  = t * 16;
        int row = n0 + l16; if (row > NTOK - 1) row = NTOK - 1;   // clamp (no divergence)
        const float* xr = xb + row * CCH;
        v8f acc = {};
        for (int j = 0; j < 48; ++j) {
            const int ch = 4 * j + 2 * half;
            v2f a;  a.x  = xr[ch];            a.y  = xr[ch + 1];
            v2f bf; bf.x = s_wdT[ch][l16];    bf.y = s_wdT[ch + 1][l16];
            acc = __builtin_amdgcn_wmma_f32_16x16x4_f32(
                false, a, false, bf, (short)0, acc, false, false);
        }
        if (l16 < DIMK) {
            const float bd = s_bd[l16];
            #pragma unroll
            for (int r = 0; r < 8; ++r) {
                const int m = n0 + r + 8 * half;
                if (m < NTOK) s_d[m][l16] = qgelu(acc[r] + bd);
            }
        }
    }
    __syncthreads();

    // ---- Phase 2: 3x3 conv over 14x14 patch grid (+cls center tap), QuickGELU ----
    for (int e = tid; e < NPAD * DIMK; e += 256) {
        const int n = e >> 3, k = e & 7;
        float o;
        if (n >= NTOK) {
            o = 0.0f;                                   // zero WMMA A pad rows
        } else if (n == 0) {                            // cls: only center tap
            float y = s_cb[k];
            const float* cw = &s_conv[k * 72 + 4];      // dh=1,dw=1
            #pragma unroll
            for (int i = 0; i < DIMK; ++i) y += cw[i * 9] * s_d[0][i];
            o = qgelu(y);
        } else if (n == 1) {
            o = 0.0f;                                   // zero_dist; qgelu(0)=0
        } else {
            const int p = n - 2, ph = p / 14, pw = p % 14;
            float y = s_cb[k];
            for (int dh = 0; dh < 3; ++dh) {
                const int hh = ph + dh - 1;
                if (hh < 0 || hh > 13) continue;
                for (int dw = 0; dw < 3; ++dw) {
                    const int ww = pw + dw - 1;
                    if (ww < 0 || ww > 13) continue;
                    const float* dn = s_d[2 + hh * 14 + ww];
                    const float* cw = &s_conv[k * 72 + dh * 3 + dw];
                    #pragma unroll
                    for (int i = 0; i < DIMK; ++i) y += cw[i * 9] * dn[i];
                }
            }
            o = qgelu(y);
        }
        s_d2[n][k] = o;
    }
    __syncthreads();

    // ---- Phase 3: up projection via V_WMMA_F32_16X16X4_F32 (K=8 -> 2 chained WMMAs) ----
    float* ob = out + (size_t)b * NTOK * CCH;
    for (int t = wid; t < MT * NT; t += 8) {
        const int mt = t / NT, nt = t % NT;
        const int n0 = mt * 16, c0 = nt * 16;
        const int ar = n0 + l16;      // A row (d2 token)
        const int bc = c0 + l16;      // B col (output channel)
        v8f acc = {};
        #pragma unroll
        for (int j = 0; j < 2; ++j) {
            const int kk = 4 * j + 2 * half;
            v2f a;  a.x  = s_d2[ar][kk];     a.y  = s_d2[ar][kk + 1];
            v2f bf; bf.x = s_wup[bc][kk];    bf.y = s_wup[bc][kk + 1];
            acc = __builtin_amdgcn_wmma_f32_16x16x4_f32(
                false, a, false, bf, (short)0, acc, false, false);
        }
        const float bu = s_bu[bc];
        #pragma unroll
        for (int r = 0; r < 8; ++r) {
            const int m = n0 + r + 8 * half;
            if (m < NTOK) ob[(size_t)m * CCH + bc] = acc[r] + bu;
        }
    }
}

extern "C" void kernel_launch(void* const* d_in, const int* in_sizes, int n_in,
                              void* d_out, int out_size, void* d_ws, size_t ws_size,
                              hipStream_t stream) {
    const float* x      = (const float*)d_in[0];
    const float* w_down = (const float*)d_in[1];
    const float* b_down = (const float*)d_in[2];
    const float* conv_w = (const float*)d_in[3];
    const float* conv_b = (const float*)d_in[4];
    const float* w_up   = (const float*)d_in[5];
    const float* b_up   = (const float*)d_in[6];
    float* out = (float*)d_out;

    const int B = 1024;
    convpass_fused<<<B, 256, 0, stream>>>(x, w_down, b_down, conv_w, conv_b,
                                          w_up, b_up, out);
    (void)in_sizes; (void)n_in; (void)out_size; (void)d_ws; (void)ws_size;
}